// NonLinearODELayer_20899310862859
// MI455X (gfx1250) — compile-verified
//
#include <hip/hip_runtime.h>
#include <utility>

// CDNA5 / gfx1250: wave32, V_WMMA_F32_16X16X4_F32.
typedef __attribute__((ext_vector_type(2))) float v2f;
typedef __attribute__((ext_vector_type(4))) float v4f;
typedef __attribute__((ext_vector_type(8))) float v8f;

#define DDIM 32
#define NPAIR 528           // D*(D+1)/2 packed symmetric pairs (contraction dim)
#define NSTEP 132           // NPAIR / 4 K-steps per WMMA chain
#define WSTRIDE 532         // padded LDS stride (floats): conflict-free b64 reads
#define ROWS_PER_BLOCK 256  // 8 waves x 32 rows

// Upper-triangle pair enumeration: p -> (i, j), i <= j, row-major.
constexpr int pair_i(int p) {
    int i = 0, off = 0;
    while (off + (DDIM - i) <= p) { off += DDIM - i; ++i; }
    return i;
}
constexpr int pair_j(int p) {
    int i = pair_i(p);
    int off = DDIM * i - i * (i - 1) / 2;
    return i + (p - off);
}

// One K-step (4 packed pairs): build A = x_i * x_j per the f32 A layout
// (lanes 0-15 hold K={0,1}, lanes 16-31 hold K={2,3}), read B from LDS,
// issue 4 WMMAs (2 row tiles x 2 col tiles).
template<int S>
__device__ __forceinline__ void qf_step(const float (&xs0)[DDIM], const float (&xs1)[DDIM],
                                        bool hi, const float* bp0, const float* bp1,
                                        v8f& a00, v8f& a01, v8f& a10, v8f& a11)
{
    constexpr int i0 = pair_i(4*S+0), j0 = pair_j(4*S+0);
    constexpr int i1 = pair_i(4*S+1), j1 = pair_j(4*S+1);
    constexpr int i2 = pair_i(4*S+2), j2 = pair_j(4*S+2);
    constexpr int i3 = pair_i(4*S+3), j3 = pair_j(4*S+3);

    v2f u0, w0, u1, w1;
    u0.x = hi ? xs0[i2] : xs0[i0];  u0.y = hi ? xs0[i3] : xs0[i1];  // folds when i's equal
    w0.x = hi ? xs0[j2] : xs0[j0];  w0.y = hi ? xs0[j3] : xs0[j1];
    u1.x = hi ? xs1[i2] : xs1[i0];  u1.y = hi ? xs1[i3] : xs1[i1];
    w1.x = hi ? xs1[j2] : xs1[j0];  w1.y = hi ? xs1[j3] : xs1[j1];
    v2f A0 = u0 * w0;                       // v_pk_mul_f32
    v2f A1 = u1 * w1;
    v2f B0 = *(const v2f*)(bp0 + 4 * S);    // ds_load_b64, all 64 banks once
    v2f B1 = *(const v2f*)(bp1 + 4 * S);
    a00 = __builtin_amdgcn_wmma_f32_16x16x4_f32(false, A0, false, B0, (short)0, a00, false, false);
    a01 = __builtin_amdgcn_wmma_f32_16x16x4_f32(false, A0, false, B1, (short)0, a01, false, false);
    a10 = __builtin_amdgcn_wmma_f32_16x16x4_f32(false, A1, false, B0, (short)0, a10, false, false);
    a11 = __builtin_amdgcn_wmma_f32_16x16x4_f32(false, A1, false, B1, (short)0, a11, false, false);
}

template<int... S>
__device__ __forceinline__ void qf_all(std::integer_sequence<int, S...>,
                                       const float (&xs0)[DDIM], const float (&xs1)[DDIM],
                                       bool hi, const float* bp0, const float* bp1,
                                       v8f& a00, v8f& a01, v8f& a10, v8f& a11)
{
    (qf_step<S>(xs0, xs1, hi, bp0, bp1, a00, a01, a10, a11), ...);
}

// out[n,k] = sum_{p<528} Ysym[n,p] * Wsym[k,p]
//   Ysym[n,p]  = x[n,i(p)] * x[n,j(p)]
//   Wsym[k,p]  = W[k,i,j] + W[k,j,i]  (i<j),  W[k,i,i]  (diag)
__global__ __launch_bounds__(256)
void bilinear_wmma_f32_sym(const float* __restrict__ x,
                           const float* __restrict__ w,
                           float* __restrict__ out)
{
    __shared__ float wlds[DDIM * WSTRIDE];   // 66.5 KB

    // ---- stage symmetrized, pair-packed weight into LDS ----
    const int tid = threadIdx.x;
    for (int pq = tid; pq < NPAIR; pq += 256) {
        int i = 0, off = 0;
        while (off + (DDIM - i) <= pq) { off += DDIM - i; ++i; }
        const int j = i + (pq - off);
#pragma unroll
        for (int k = 0; k < DDIM; ++k) {
            const float a = w[k * (DDIM * DDIM) + i * DDIM + j];
            const float b = w[k * (DDIM * DDIM) + j * DDIM + i];
            wlds[k * WSTRIDE + pq] = (i == j) ? a : (a + b);
        }
    }
    __syncthreads();

    const int lane = tid & 31;
    const int wave = tid >> 5;
    const int l15  = lane & 15;
    const bool hi  = lane >= 16;

    const int wbase = blockIdx.x * ROWS_PER_BLOCK + wave * 32;
    const int m0 = wbase + l15;          // row tile 0 (A layout: lane%16 = M)
    const int m1 = m0 + 16;              // row tile 1

    // ---- stage the two x rows this lane needs into registers ----
    float xs0[DDIM], xs1[DDIM];
#pragma unroll
    for (int q = 0; q < 8; ++q) {
        v4f a = *(const v4f*)(x + m0 * DDIM + 4 * q);
        v4f b = *(const v4f*)(x + m1 * DDIM + 4 * q);
        xs0[4*q+0] = a.x; xs0[4*q+1] = a.y; xs0[4*q+2] = a.z; xs0[4*q+3] = a.w;
        xs1[4*q+0] = b.x; xs1[4*q+1] = b.y; xs1[4*q+2] = b.z; xs1[4*q+3] = b.w;
    }

    v8f acc00 = {}, acc01 = {}, acc10 = {}, acc11 = {};

    // B lane base: Wsym[16c + l15, 4S + 2*hi (+1)]
    const float* bp0 = &wlds[l15 * WSTRIDE + (hi ? 2 : 0)];
    const float* bp1 = bp0 + 16 * WSTRIDE;

    qf_all(std::make_integer_sequence<int, NSTEP>{},
           xs0, xs1, hi, bp0, bp1, acc00, acc01, acc10, acc11);

    // ---- D layout: lane%16 = N column, VGPR r = row r (+8 for upper lanes) ----
#pragma unroll
    for (int r = 0; r < 8; ++r) {
        const int row0 = wbase + r + (hi ? 8 : 0);
        out[row0 * DDIM + l15]             = acc00[r];
        out[row0 * DDIM + 16 + l15]        = acc01[r];
        out[(row0 + 16) * DDIM + l15]      = acc10[r];
        out[(row0 + 16) * DDIM + 16 + l15] = acc11[r];
    }
}

extern "C" void kernel_launch(void* const* d_in, const int* in_sizes, int n_in,
                              void* d_out, int out_size, void* d_ws, size_t ws_size,
                              hipStream_t stream) {
    (void)n_in; (void)out_size; (void)d_ws; (void)ws_size;
    const float* x = (const float*)d_in[0];   // (N, 32) f32
    const float* w = (const float*)d_in[1];   // (32, 32, 32) f32
    float* out = (float*)d_out;               // (N, 32) f32

    const int rows = in_sizes[0] / DDIM;      // 262144
    dim3 grid(rows / ROWS_PER_BLOCK);         // 1024
    dim3 block(256);                          // 8 wave32 waves
    hipLaunchKernelGGL(bilinear_wmma_f32_sym, grid, block, 0, stream, x, w, out);
}